// VolumeRenderer_64192581206347
// MI455X (gfx1250) — compile-verified
//
#include <hip/hip_runtime.h>
#include <hip/hip_bf16.h>
#include <stdint.h>

#define BIG_DELTA 1e10f
#define NPTS 192
#define CHUNKS 6            // 192 / 32 lanes
#define WAVES_PER_BLOCK 8
#define RAY_BYTES 3840      // 768 (depth) + 768 (density) + 2304 (feature)
#define DEPTH_OFF 0
#define DENS_OFF  768
#define FEAT_OFF  1536

// One wave-wide async transfer: 16 B per active lane, LDS and global addresses
// both advanced by the 24-bit instruction immediate (ISA: INST_OFFSET is added
// to both sides). Tracked by ASYNCcnt.
#define ASYNC_LD_B128(l, g, OFF)                                        \
  asm volatile("global_load_async_to_lds_b128 %0, %1, off offset:%2"    \
               :: "v"(l), "v"(g), "i"(OFF) : "memory")

__device__ __forceinline__ void wait_async_zero() {
#if __has_builtin(__builtin_amdgcn_s_wait_asynccnt)
  __builtin_amdgcn_s_wait_asynccnt(0);
#else
  asm volatile("s_wait_asynccnt 0x0" ::: "memory");
#endif
  asm volatile("" ::: "memory");  // compiler-level fence: LDS now populated
}

__global__ __launch_bounds__(256) void VolumeRenderer_kernel(
    const float* __restrict__ depth, const float* __restrict__ density,
    const float* __restrict__ feature, float* __restrict__ out, int R) {
  __shared__ __align__(16) unsigned char smem[WAVES_PER_BLOCK * RAY_BYTES];

  const int lane = threadIdx.x & 31;
  const int wave = threadIdx.x >> 5;
  const int r = blockIdx.x * WAVES_PER_BLOCK + wave;
  if (r >= R) return;

  // Low 32 bits of the flat shared-aperture address == wave-relative LDS byte
  // offset (ISA 10.2 aperture mapping), which is what the async VDST wants.
  const uint32_t lds_base = (uint32_t)(uintptr_t)&smem[wave * RAY_BYTES];
  const uint32_t lane16   = (uint32_t)(lane * 16);

  const uint32_t l_dep = lds_base + DEPTH_OFF + lane16;
  const uint32_t l_den = lds_base + DENS_OFF  + lane16;
  const uint32_t l_fea = lds_base + FEAT_OFF  + lane16;
  const uint64_t g_dep = (uint64_t)(uintptr_t)(depth   + (size_t)r * NPTS)     + lane16;
  const uint64_t g_den = (uint64_t)(uintptr_t)(density + (size_t)r * NPTS)     + lane16;
  const uint64_t g_fea = (uint64_t)(uintptr_t)(feature + (size_t)r * NPTS * 3) + lane16;

  // depth: 768 B = 512 (all lanes) + 256 (lanes 0-15)
  ASYNC_LD_B128(l_dep, g_dep, 0);
  // density: 768 B
  ASYNC_LD_B128(l_den, g_den, 0);
  // feature: 2304 B = 4*512 (all lanes) + 256 (lanes 0-15)
  ASYNC_LD_B128(l_fea, g_fea, 0);
  ASYNC_LD_B128(l_fea, g_fea, 512);
  ASYNC_LD_B128(l_fea, g_fea, 1024);
  ASYNC_LD_B128(l_fea, g_fea, 1536);
  if (lane < 16) {
    ASYNC_LD_B128(l_dep, g_dep, 512);
    ASYNC_LD_B128(l_den, g_den, 512);
    ASYNC_LD_B128(l_fea, g_fea, 2048);
  }
  wait_async_zero();

  const float* sdep = (const float*)(smem + wave * RAY_BYTES + DEPTH_OFF);
  const float* sden = (const float*)(smem + wave * RAY_BYTES + DENS_OFF);
  const float* sfea = (const float*)(smem + wave * RAY_BYTES + FEAT_OFF);

  float carry = 0.0f;          // running cumsum of tm up to previous chunk
  float ar = 0.0f, ag = 0.0f, ab = 0.0f, ad = 0.0f;

#pragma unroll
  for (int c = 0; c < CHUNKS; ++c) {
    const int p = c * 32 + lane;            // lane-major: coalesced & conflict-free
    const float d  = sdep[p];
    const float sg = sden[p];
    const float f0 = sfea[3 * p + 0];
    const float f1 = sfea[3 * p + 1];
    const float f2 = sfea[3 * p + 2];

    const bool  last  = (p == NPTS - 1);
    const float dn    = last ? d : sdep[p + 1];
    const float delta = last ? BIG_DELTA : (dn - d);
    const float tm    = delta * sg;

    // inclusive wave32 scan of tm
    float scan = tm;
#pragma unroll
    for (int o = 1; o < 32; o <<= 1) {
      float y = __shfl_up(scan, o, 32);
      if (lane >= o) scan += y;
    }

    const float cum = carry + scan;         // inclusive cumsum incl. this point
    const float T   = __expf(-cum);         // transmittance
    const float w   = T * (1.0f - __expf(-tm));

    ar += w * f0;
    ag += w * f1;
    ab += w * f2;
    ad += w * d;

    carry += __shfl(scan, 31, 32);          // chunk total -> carry
  }

  // butterfly reduction across the wave for the 4 accumulators
#pragma unroll
  for (int o = 16; o >= 1; o >>= 1) {
    ar += __shfl_xor(ar, o, 32);
    ag += __shfl_xor(ag, o, 32);
    ab += __shfl_xor(ab, o, 32);
    ad += __shfl_xor(ad, o, 32);
  }

  if (lane == 0) {
    float4 v = make_float4(ar, ag, ab, ad);
    *(float4*)(out + (size_t)r * 4) = v;
  }
}

extern "C" void kernel_launch(void* const* d_in, const int* in_sizes, int n_in,
                              void* d_out, int out_size, void* d_ws, size_t ws_size,
                              hipStream_t stream) {
  const float* depth   = (const float*)d_in[0];   // (R, 192)
  const float* density = (const float*)d_in[1];   // (R, 192, 1)
  const float* feature = (const float*)d_in[2];   // (R, 192, 3)
  float* out = (float*)d_out;                     // (R, 4)

  const int R = in_sizes[0] / NPTS;
  const int blocks = (R + WAVES_PER_BLOCK - 1) / WAVES_PER_BLOCK;
  hipLaunchKernelGGL(VolumeRenderer_kernel, dim3(blocks), dim3(256), 0, stream,
                     depth, density, feature, out, R);
}